// GAT_75084618269291
// MI455X (gfx1250) — compile-verified
//
#include <hip/hip_runtime.h>
#include <hip/hip_bf16.h>
#include <math.h>

#define H 8
#define C 10
#define DOUT 80           // H*C
#define NEG_SLOPE 0.2f

typedef __attribute__((ext_vector_type(2))) float v2f;
typedef __attribute__((ext_vector_type(8))) float v8f;

// ---------------------------------------------------------------------------
// monotone float <-> ordered-uint bijection for atomic float max
// ---------------------------------------------------------------------------
__device__ __forceinline__ unsigned f2ord(float f) {
    unsigned u = __float_as_uint(f);
    return (u & 0x80000000u) ? ~u : (u | 0x80000000u);
}
__device__ __forceinline__ float ord2f(unsigned u) {
    return (u & 0x80000000u) ? __uint_as_float(u & 0x7fffffffu)
                             : __uint_as_float(~u);
}
// f2ord(-inf) == 0x007fffff
#define ORD_NEG_INF 0x007fffffu

// ---------------------------------------------------------------------------
// repack W (din x 80, row-major) into K-pair-interleaved float2 layout:
//   Wp[p*80 + c] = { W[2p][c], W[2p+1][c] }   for p in [0, din/2)
// so a WMMA B-fragment is a single coalesced 8-byte load.
// ---------------------------------------------------------------------------
__global__ void gat_repack_w(const float* __restrict__ Wl,
                             const float* __restrict__ Wr,
                             float2* __restrict__ Wlp,
                             float2* __restrict__ Wrp,
                             int din)
{
    const int idx = blockIdx.x * blockDim.x + threadIdx.x;   // p*80 + c
    const int total = (din >> 1) * DOUT;
    if (idx >= total) return;
    const int p = idx / DOUT;
    const int c = idx - p * DOUT;
    float2 l, r;
    l.x = Wl[(size_t)(2 * p) * DOUT + c];
    l.y = Wl[(size_t)(2 * p + 1) * DOUT + c];
    r.x = Wr[(size_t)(2 * p) * DOUT + c];
    r.y = Wr[(size_t)(2 * p + 1) * DOUT + c];
    Wlp[idx] = l;
    Wrp[idx] = r;
}

// ---------------------------------------------------------------------------
// WMMA GEMM: xl = x @ Wl, xr = x @ Wr   (x: n x DIN row-major, W pre-packed)
// One wave32 per 16-row node tile; 5 column tiles x 2 weight matrices
// = 10 f32 16x16 accumulators in VGPRs; V_WMMA_F32_16X16X4_F32 stepping K by 4.
// DIN is a compile-time constant so the fully-unrolled K loop reduces to
// immediate-offset b64 loads from three per-lane base pointers; explicit
// fragment arrays force batched loads ahead of the WMMA group.
// ---------------------------------------------------------------------------
template <int DIN>
__global__ __launch_bounds__(128)
void gat_gemm_wmma(const float* __restrict__ x,
                   const float2* __restrict__ Wlp,
                   const float2* __restrict__ Wrp,
                   float* __restrict__ xl,
                   float* __restrict__ xr,
                   int n)
{
    const int lane  = threadIdx.x & 31;
    const int wave  = threadIdx.x >> 5;
    const int tileM = blockIdx.x * 4 + wave;          // wave-uniform
    if (tileM * 16 >= n) return;                      // whole-wave exit

    const int rowA = tileM * 16 + (lane & 15);        // A: M = lane&15
    const int koff = (lane >> 4) << 1;                // A/B: K sub-offset 0 or 2
    const int colB = lane & 15;                       // B: N = lane&15

    v8f accL[5], accR[5];
#pragma unroll
    for (int t = 0; t < 5; ++t)
#pragma unroll
        for (int i = 0; i < 8; ++i) { accL[t][i] = 0.f; accR[t][i] = 0.f; }

    // per-lane base pointers; all K-loop addressing is immediate offsets
    const float2* aBase  = (const float2*)(x + (size_t)rowA * DIN + koff);
    const float2* wlBase = Wlp + (size_t)(koff >> 1) * DOUT + colB;
    const float2* wrBase = Wrp + (size_t)(koff >> 1) * DOUT + colB;

#pragma unroll
    for (int k = 0; k < DIN; k += 4) {
        const int prow = k >> 1;                      // packed K-pair row
        // batch-issue A + 10 B fragment loads before any WMMA
        float2 a2 = aBase[k >> 1];
        float2 lv[5], rv[5];
#pragma unroll
        for (int t = 0; t < 5; ++t) {
            lv[t] = wlBase[prow * DOUT + t * 16];
            rv[t] = wrBase[prow * DOUT + t * 16];
        }
        v2f a; a.x = a2.x; a.y = a2.y;
#pragma unroll
        for (int t = 0; t < 5; ++t) {
            v2f bl; bl.x = lv[t].x; bl.y = lv[t].y;
            v2f br; br.x = rv[t].x; br.y = rv[t].y;
            accL[t] = __builtin_amdgcn_wmma_f32_16x16x4_f32(
                false, a, false, bl, (short)0, accL[t], false, false);
            accR[t] = __builtin_amdgcn_wmma_f32_16x16x4_f32(
                false, a, false, br, (short)0, accR[t], false, false);
        }
    }

    // C/D layout: VGPR i -> row (lane<16 ? i : 8+i), col = lane&15
    const int rbase = tileM * 16 + ((lane >> 4) << 3);
#pragma unroll
    for (int t = 0; t < 5; ++t) {
        const int c = t * 16 + colB;
#pragma unroll
        for (int i = 0; i < 8; ++i) {
            xl[(size_t)(rbase + i) * DOUT + c] = accL[t][i];
            xr[(size_t)(rbase + i) * DOUT + c] = accR[t][i];
        }
    }
}

// ---------------------------------------------------------------------------
// per-layer init: accum = 0, smax = ord(-inf), denom = 0
// ---------------------------------------------------------------------------
__global__ void gat_init(float* __restrict__ accum,
                         unsigned* __restrict__ smax,
                         float* __restrict__ denom, int n)
{
    const int idx = blockIdx.x * blockDim.x + threadIdx.x;
    if (idx < n * DOUT) accum[idx] = 0.f;
    if (idx < n * H)   { smax[idx] = ORD_NEG_INF; denom[idx] = 0.f; }
}

// ---------------------------------------------------------------------------
// edge pass 1: score[e,h] = att[h] . leaky_relu(xl[src]+xr[dst]); seg-max
// ---------------------------------------------------------------------------
__global__ void gat_score(const long long* __restrict__ ei, int Eraw, int Etot,
                          const float* __restrict__ xl,
                          const float* __restrict__ xr,
                          const float* __restrict__ att,
                          float* __restrict__ scores,
                          unsigned* __restrict__ smax)
{
    const int idx = blockIdx.x * blockDim.x + threadIdx.x;
    if (idx >= Etot * H) return;
    const int e = idx >> 3, h = idx & 7;
    int src, dst;
    if (e < Eraw) { src = (int)ei[e]; dst = (int)ei[e + Eraw]; }
    else          { src = dst = e - Eraw; }                     // self loop

    const float2* pl = (const float2*)(xl + (size_t)src * DOUT + h * C);
    const float2* pr = (const float2*)(xr + (size_t)dst * DOUT + h * C);
    const float2* pa = (const float2*)(att + h * C);
    float s = 0.f;
#pragma unroll
    for (int c = 0; c < C / 2; ++c) {
        const float2 l = pl[c], r = pr[c], a = pa[c];
        float v0 = l.x + r.x;
        float v1 = l.y + r.y;
        v0 = (v0 > 0.f) ? v0 : NEG_SLOPE * v0;
        v1 = (v1 > 0.f) ? v1 : NEG_SLOPE * v1;
        s += v0 * a.x + v1 * a.y;
    }
    scores[idx] = s;
    atomicMax(smax + (size_t)dst * H + h, f2ord(s));
}

// ---------------------------------------------------------------------------
// edge pass 2: ex = exp(score - smax[dst]); seg-sum into denom
// ---------------------------------------------------------------------------
__global__ void gat_expsum(const long long* __restrict__ ei, int Eraw, int Etot,
                           float* __restrict__ scores,
                           const unsigned* __restrict__ smax,
                           float* __restrict__ denom)
{
    const int idx = blockIdx.x * blockDim.x + threadIdx.x;
    if (idx >= Etot * H) return;
    const int e = idx >> 3, h = idx & 7;
    const int dst = (e < Eraw) ? (int)ei[e + Eraw] : e - Eraw;
    const float m  = ord2f(smax[(size_t)dst * H + h]);
    const float ex = expf(scores[idx] - m);
    scores[idx] = ex;
    atomicAdd(denom + (size_t)dst * H + h, ex);
}

// ---------------------------------------------------------------------------
// edge pass 3: alpha = ex/denom; accum[dst] += alpha * xl[src]
// ---------------------------------------------------------------------------
__global__ void gat_scatter(const long long* __restrict__ ei, int Eraw, int Etot,
                            const float* __restrict__ scores,
                            const float* __restrict__ denom,
                            const float* __restrict__ xl,
                            float* __restrict__ accum)
{
    const int idx = blockIdx.x * blockDim.x + threadIdx.x;
    if (idx >= Etot * H) return;
    const int e = idx >> 3, h = idx & 7;
    int src, dst;
    if (e < Eraw) { src = (int)ei[e]; dst = (int)ei[e + Eraw]; }
    else          { src = dst = e - Eraw; }

    const float alpha = scores[idx] / (denom[(size_t)dst * H + h] + 1e-16f);
    const float2* pl = (const float2*)(xl + (size_t)src * DOUT + h * C);
    float*        po = accum + (size_t)dst * DOUT + h * C;
#pragma unroll
    for (int c = 0; c < C / 2; ++c) {
        const float2 l = pl[c];
        atomicAdd(po + 2 * c + 0, alpha * l.x);
        atomicAdd(po + 2 * c + 1, alpha * l.y);
    }
}

// ---------------------------------------------------------------------------
// epilogue: out = elu(accum + bias)
// ---------------------------------------------------------------------------
__global__ void gat_finish(const float* __restrict__ accum,
                           const float* __restrict__ bias,
                           float* __restrict__ out, int n)
{
    const int idx = blockIdx.x * blockDim.x + threadIdx.x;
    if (idx >= n * DOUT) return;
    const float v = accum[idx] + bias[idx % DOUT];
    out[idx] = (v > 0.f) ? v : expm1f(v);
}

// ---------------------------------------------------------------------------
extern "C" void kernel_launch(void* const* d_in, const int* in_sizes, int n_in,
                              void* d_out, int out_size, void* d_ws, size_t ws_size,
                              hipStream_t stream)
{
    const float*     x0 = (const float*)d_in[0];
    const long long* ei = (const long long*)d_in[1];   // int64 edge_index [2,E]

    const int n    = in_sizes[0] / 128;                // 50000
    const int Eraw = in_sizes[1] / 2;                  // 1.6M
    const int Etot = Eraw + n;                         // + self loops

    // carve workspace (256B aligned slices)
    char* p = (char*)d_ws;
    auto carve = [&](size_t bytes) -> void* {
        void* r = (void*)p;
        p += (bytes + 255) & ~(size_t)255;
        return r;
    };
    float*    xl     = (float*)   carve((size_t)n * DOUT * sizeof(float));
    float*    xr     = (float*)   carve((size_t)n * DOUT * sizeof(float));
    float*    accum  = (float*)   carve((size_t)n * DOUT * sizeof(float));
    float*    xbuf   = (float*)   carve((size_t)n * DOUT * sizeof(float));
    unsigned* smax   = (unsigned*)carve((size_t)n * H    * sizeof(unsigned));
    float*    denom  = (float*)   carve((size_t)n * H    * sizeof(float));
    float*    scores = (float*)   carve((size_t)Etot * H * sizeof(float));
    float2*   Wlp    = (float2*)  carve((size_t)64 * DOUT * sizeof(float2)); // max din/2
    float2*   Wrp    = (float2*)  carve((size_t)64 * DOUT * sizeof(float2));

    const int tiles     = (n + 15) / 16;
    const int gemmBlk   = (tiles + 3) / 4;                 // 4 waves / block
    const int initThr   = n * DOUT;
    const int initBlk   = (initThr + 255) / 256;
    const int edgeThr   = Etot * H;
    const int edgeBlk   = (edgeThr + 255) / 256;
    const int finBlk    = (n * DOUT + 255) / 256;

    const float* xin = x0;
    for (int layer = 0; layer < 3; ++layer) {
        const float* Wl  = (const float*)d_in[2 + layer * 4 + 0];
        const float* Wr  = (const float*)d_in[2 + layer * 4 + 1];
        const float* att = (const float*)d_in[2 + layer * 4 + 2];
        const float* b   = (const float*)d_in[2 + layer * 4 + 3];
        float* out = (layer == 2) ? (float*)d_out : xbuf;
        const int din = (layer == 0) ? 128 : DOUT;

        const int packThr = (din >> 1) * DOUT;
        gat_repack_w<<<(packThr + 255) / 256, 256, 0, stream>>>(Wl, Wr, Wlp, Wrp, din);
        if (layer == 0)
            gat_gemm_wmma<128><<<gemmBlk, 128, 0, stream>>>(xin, Wlp, Wrp, xl, xr, n);
        else
            gat_gemm_wmma<DOUT><<<gemmBlk, 128, 0, stream>>>(xin, Wlp, Wrp, xl, xr, n);
        gat_init    <<<initBlk, 256, 0, stream>>>(accum, smax, denom, n);
        gat_score   <<<edgeBlk, 256, 0, stream>>>(ei, Eraw, Etot, xl, xr, att,
                                                  scores, smax);
        gat_expsum  <<<edgeBlk, 256, 0, stream>>>(ei, Eraw, Etot, scores, smax,
                                                  denom);
        gat_scatter <<<edgeBlk, 256, 0, stream>>>(ei, Eraw, Etot, scores, denom,
                                                  xl, accum);
        gat_finish  <<<finBlk, 256, 0, stream>>>(accum, b, out, n);

        xin = out;
    }
}